// ModelNew_67276367724715
// MI455X (gfx1250) — compile-verified
//
#include <hip/hip_runtime.h>
#include <hip/hip_bf16.h>

typedef __attribute__((ext_vector_type(2))) float v2f;
typedef __attribute__((ext_vector_type(8))) float v8f;

// Pointer types for __builtin_amdgcn_global_load_async_to_lds_b32:
//   param0: i32 addrspace(1)* (global), param1: i32 addrspace(3)* (LDS)
typedef __attribute__((address_space(1))) int* gptr_b32;
typedef __attribute__((address_space(3))) int* lptr_b32;

#define CIN    128
#define COUT   256
#define HIN    64
#define WIN    64
#define OHW    62            // output H and W (valid 3x3 conv)
#define KSZ    (CIN * 9)     // 1152 = GEMM K
#define CCH    8             // cin chunk per LDS round
#define NCH    (CIN / CCH)   // 16 chunks
#define XSTR2  68            // LDS row stride in float2 units: 64 data + 4 pad
#define XBUF   (4 * 3 * XSTR2 * 2)   // floats per buffer (cp, row, col, pair)
#define NKP    (KSZ / 2)     // 576 packed K-pairs in wT
#define GELU_K0 0.7978845608028654f

// ---- CDNA5 async global->LDS path (guarded; sync fallback otherwise) ------
#if __has_builtin(__builtin_amdgcn_global_load_async_to_lds_b32)
#define HAS_ASYNC 1
#else
#define HAS_ASYNC 0
#endif

#if __has_builtin(__builtin_amdgcn_s_wait_asynccnt)
#define WAIT_ASYNC() __builtin_amdgcn_s_wait_asynccnt(0)
#else
#define WAIT_ASYNC() asm volatile("s_wait_asynccnt 0x0" ::: "memory")
#endif

// ---------------------------------------------------------------------------
__global__ void zero_out_kernel(float* __restrict__ out, int n) {
    int i = blockIdx.x * blockDim.x + threadIdx.x;
    if (i < n) out[i] = 0.0f;
}

// ---------------------------------------------------------------------------
// Pack weights K-major for coalesced A-fragment loads:
//   kk = (c*9 + khw)*8 + cl   (c = cin chunk, khw = tap, cl = cin-in-chunk)
//   wt[(p*COUT + cout)*2 + {0,1}] = wgt[cout][kk=2p], wgt[cout][kk=2p+1]
// ---------------------------------------------------------------------------
__global__ void wt_pack_kernel(const float* __restrict__ wgt, float* __restrict__ wt) {
    int idx = blockIdx.x * blockDim.x + threadIdx.x;
    if (idx >= NKP * COUT) return;
    int p   = idx >> 8;          // K pair 0..575
    int j   = idx & 255;         // cout
    int kk0 = p * 2;
    int c   = kk0 / 72;          // 72 K values per cin chunk
    int rem = kk0 - c * 72;
    int khw = rem >> 3;
    int cl  = rem & 7;           // even
    int cin = c * CCH + cl;
    const float* wr = wgt + (size_t)j * KSZ;
    wt[2 * ((size_t)p * COUT + j) + 0] = wr[(cin    ) * 9 + khw];
    wt[2 * ((size_t)p * COUT + j) + 1] = wr[(cin + 1) * 9 + khw];
}

// ---------------------------------------------------------------------------
// Stage one cin-chunk of x (8 cins x 3 rows x 64 cols) into LDS with
// adjacent-cin pairs interleaved:  LDS float2[cp][row][col] = {cin 2cp, 2cp+1}.
// Per-lane 4B async scatter (ASYNCcnt); global side stays coalesced.
// Pad columns 64..67 are never written: they feed only masked outputs.
// ---------------------------------------------------------------------------
__device__ __forceinline__ void stage_x(const float* __restrict__ xb, int c, int oh,
                                        float* __restrict__ dst, int tid) {
    #pragma unroll
    for (int i = 0; i < 12; ++i) {
        int idx   = tid + 128 * i;            // 0..1535
        int cin_l = idx / 192;                // 192 = 3 rows * 64 cols
        int rem   = idx - cin_l * 192;
        int r     = rem >> 6;
        int col   = rem & 63;
        const float* g = xb + (size_t)(c * CCH + cin_l) * (HIN * WIN)
                            + (oh + r) * WIN + col;
        // float index: pair-interleaved
        float* l = dst + (((cin_l >> 1) * 3 + r) * XSTR2 + col) * 2 + (cin_l & 1);
#if HAS_ASYNC
        __builtin_amdgcn_global_load_async_to_lds_b32(
            (gptr_b32)g, (lptr_b32)l, 0, 0);
#else
        *l = *g;
#endif
    }
}

// ---------------------------------------------------------------------------
// Fused conv3x3(valid) + bias + tanh-GELU + global average pool.
// grid = (62 rows, 4 cout blocks of 64, 32 batches); block = 128 = 4 wave32.
// Wave w owns couts [cout0+16w, +16) x all 64 columns -> 4 f32 accumulators.
// Per K-step: 1 global b64 (A, K-major packed) + 4 ds_load_b64 (B pairs)
//             + 4 independent v_wmma_f32_16x16x4_f32 (no repack moves).
// ---------------------------------------------------------------------------
template <bool USE_WT>
__global__ __launch_bounds__(128) void conv_gelu_pool_wmma(
    const float* __restrict__ x,     // (32,128,64,64)
    const float* __restrict__ wA,    // USE_WT ? packed wT : raw weights
    const float* __restrict__ bias,  // (256,)
    float* __restrict__ out)         // (32,256)
{
    __shared__ float lds_x[2 * XBUF];   // double-buffered x tile

    const int tid  = threadIdx.x;
    const int lane = tid & 31;
    const int wv   = tid >> 5;        // wave 0..3
    const int hi   = lane >> 4;       // half-wave -> K-pair select (ISA layout)
    const int n    = lane & 15;       // M (for A) / N (for B,C) index

    const int oh    = blockIdx.x;          // 0..61
    const int cout0 = blockIdx.y * 64;     // cout block of 64
    const int b     = blockIdx.z;

    const float* xb    = x + (size_t)b * CIN * (HIN * WIN);
    const int    coutg = cout0 + 16 * wv + n;   // this lane's A row

    v8f acc[4] = {v8f{}, v8f{}, v8f{}, v8f{}};

    stage_x(xb, 0, oh, lds_x, tid);
#if HAS_ASYNC
    WAIT_ASYNC();
#endif
    __syncthreads();

    for (int c = 0; c < NCH; ++c) {
        const v2f* cur2 = (const v2f*)(lds_x + (c & 1) * XBUF);
        if (c + 1 < NCH)
            stage_x(xb, c + 1, oh, lds_x + ((c + 1) & 1) * XBUF, tid);

        #pragma unroll
        for (int khw = 0; khw < 9; ++khw) {
            const int kh = khw / 3;
            const int kw = khw - 3 * kh;
            #pragma unroll
            for (int s = 0; s < 2; ++s) {
                // ---- A fragment: lanes 0-15 K pair (0,1), lanes 16-31 (2,3)
                v2f a;
                if (USE_WT) {
                    const int p = (c * 9 + khw) * 4 + s * 2 + hi;  // packed K pair
                    a = *(const v2f*)(wA + 2 * ((size_t)p * COUT + coutg));
                } else {
                    const int cin = c * CCH + s * 4 + 2 * hi;
                    const float* ap = wA + (size_t)coutg * KSZ + cin * 9 + khw;
                    a.x = ap[0];
                    a.y = ap[9];
                }
                // ---- B fragments: cin pair cp = s*2 + hi, column = 16t + n + kw
                const int cp = s * 2 + hi;
                const v2f* bx = cur2 + (cp * 3 + kh) * XSTR2 + kw + n;
                #pragma unroll
                for (int t = 0; t < 4; ++t) {
                    v2f bb = bx[16 * t];       // single ds_load_b64, aligned pair
                    acc[t] = __builtin_amdgcn_wmma_f32_16x16x4_f32(
                                 false, a, false, bb, (short)0, acc[t],
                                 false, false);
                }
            }
        }

#if HAS_ASYNC
        if (c + 1 < NCH) WAIT_ASYNC();
#endif
        __syncthreads();
    }

    // ---- epilogue: bias + tanh-GELU, mask cols >= 62, pool, accumulate ----
    const float scale = 1.0f / (62.0f * 62.0f);
    #pragma unroll
    for (int r = 0; r < 8; ++r) {
        const int cg = cout0 + 16 * wv + r + 8 * hi;   // C layout: M = r + 8*hi
        const float bv = bias[cg];
        float g = 0.0f;
        #pragma unroll
        for (int t = 0; t < 4; ++t) {
            const int ow = 16 * t + n;
            float v  = acc[t][r] + bv;
            float u  = GELU_K0 * (v + 0.044715f * v * v * v);
            float th = 1.0f - 2.0f / (1.0f + __expf(2.0f * u));   // tanh(u)
            float ge = 0.5f * v * (1.0f + th);
            g += (ow < OHW) ? ge : 0.0f;    // select, so pad garbage never sums
        }
        // reduce the 16 lanes of each half-wave (column reduction)
        g += __shfl_xor(g, 1, 32);
        g += __shfl_xor(g, 2, 32);
        g += __shfl_xor(g, 4, 32);
        g += __shfl_xor(g, 8, 32);
        if (n == 0)
            unsafeAtomicAdd(&out[b * COUT + cg], g * scale);
    }
}

// ---------------------------------------------------------------------------
extern "C" void kernel_launch(void* const* d_in, const int* in_sizes, int n_in,
                              void* d_out, int out_size, void* d_ws, size_t ws_size,
                              hipStream_t stream) {
    const float* x    = (const float*)d_in[0];
    const float* wgt  = (const float*)d_in[1];
    const float* bias = (const float*)d_in[2];
    float* out = (float*)d_out;
    float* wt  = (float*)d_ws;

    zero_out_kernel<<<(out_size + 255) / 256, 256, 0, stream>>>(out, out_size);

    const size_t wt_bytes = (size_t)NKP * COUT * 2 * sizeof(float);
    dim3 grid(OHW, COUT / 64, 32);
    if (ws_size >= wt_bytes) {
        wt_pack_kernel<<<(NKP * COUT + 255) / 256, 256, 0, stream>>>(wgt, wt);
        conv_gelu_pool_wmma<true><<<grid, dim3(128), 0, stream>>>(x, wt, bias, out);
    } else {
        conv_gelu_pool_wmma<false><<<grid, dim3(128), 0, stream>>>(x, wgt, bias, out);
    }
}